// WaveletKANClassifier_76922864271551
// MI455X (gfx1250) — compile-verified
//
#include <hip/hip_runtime.h>
#include <math.h>

typedef __attribute__((ext_vector_type(2))) float v2f;
typedef __attribute__((ext_vector_type(4))) float v4f;
typedef __attribute__((ext_vector_type(8))) float v8f;

#define NROWS   32768
#define DIN     768
#define DOUT    768
#define NW      48
#define M_TILE  32    // 2 M-subtiles of 16; halves per-block W L2 traffic, B-frag reuse x2
#define XS_LD   772   // 768 + 4 pad: lane m -> bank 4m+k, all 64 banks hit once per b64 A load
#define WV_LD   52    // 48 + 4 pad
#define LN_EPS  1e-5f

__device__ __forceinline__ v8f wmma4(v2f a, v2f b, v8f c) {
  // D = A(16x4,f32) x B(4x16,f32) + C(16x16,f32): exact fp32 on the matrix pipe
  return __builtin_amdgcn_wmma_f32_16x16x4_f32(false, a, false, b, (short)0, c,
                                               false, false);
}

__global__ __launch_bounds__(256)
void wkan_fused_kernel(const float* __restrict__ x,   const float* __restrict__ W,
                       const float* __restrict__ b,   const float* __restrict__ Wp,
                       const float* __restrict__ bp,  const float* __restrict__ Wc,
                       const float* __restrict__ bc,  const float* __restrict__ scales,
                       const float* __restrict__ trans, const float* __restrict__ gamma,
                       const float* __restrict__ beta, float* __restrict__ out)
{
  __shared__ float xs[M_TILE * XS_LD];    // ~98.8 KB  x tile (32 x 768)
  __shared__ float wavs[M_TILE * WV_LD];  // ~6.7 KB   gelu(wavelet) tile (32 x 48)
  __shared__ float psum[8][M_TILE];       // per-wave row partial sums
  __shared__ float psq[8][M_TILE];        // per-wave row partial sum-of-squares
  __shared__ float s_mu[M_TILE];
  __shared__ float s_inv[M_TILE];

  const int tid  = threadIdx.x;
  const int lane = tid & 31;
  const int wid  = tid >> 5;    // 8 waves
  const int lm   = lane & 15;   // N (and A-row) index within a 16-wide tile
  const int half = lane >> 4;   // K-half selector for A/B fragments
  const int m0   = blockIdx.x * M_TILE;

  // ---- Phase A: stage x tile (32 x 768 fp32) into LDS; NT loads (x is read once) ----
  {
    const float* xg = x + (size_t)m0 * DIN;
    for (int i = tid; i < M_TILE * (DIN / 4); i += 256) {
      int row = i / (DIN / 4);
      int c4  = i % (DIN / 4);
      v4f v = __builtin_nontemporal_load((const v4f*)(xg + row * DIN + c4 * 4));
      *(v4f*)&xs[row * XS_LD + c4 * 4] = v;
    }
  }
  __syncthreads();

  // ---- Phase B: waves 0..5 compute wi = x@Wp^T (6 tiles = 2 Msub x 3 families),
  //      then wavelet basis + exact gelu -> wavs in LDS. Family uniform per wave. ----
  if (wid < 6) {
    const int msub = wid / 3;                 // M-subtile (rows 0-15 or 16-31)
    const int jt   = wid % 3;                 // wavelet family tile
    const int j    = jt * 16 + lm;            // wavelet index 0..47
    const float* wprow = Wp + (size_t)j * DIN;
    const float* arow  = &xs[(msub * 16 + lm) * XS_LD];
    v8f cwi = {};
    for (int k = 0; k < DIN; k += 4) {
      v2f a  = *(const v2f*)&arow[k + 2 * half];
      v2f bb = *(const v2f*)&wprow[k + 2 * half];
      cwi = wmma4(a, bb, cwi);
    }
    const float inv_sc = 1.0f / scales[j];
    const float tr     = trans[j];
    const float bpj    = bp[j];
    #pragma unroll
    for (int r = 0; r < 8; ++r) {
      int   m = r + 8 * half + 16 * msub;
      float s = (cwi[r] + bpj - tr) * inv_sc;
      float w;
      if (jt == 0) {        // haar
        w = (s >= 0.f && s < 0.5f) ? 1.f : ((s >= 0.5f && s < 1.f) ? -1.f : 0.f);
      } else if (jt == 1) { // mexican hat
        w = (1.f - s * s) * expf(-0.5f * s * s);
      } else {              // morlet, OMEGA0 = 5
        w = cosf(5.f * s) * expf(-0.5f * s * s);
      }
      // exact gelu: 0.5*w*(1+erf(w/sqrt(2)))
      wavs[m * WV_LD + j] = 0.5f * w * (1.f + erff(w * 0.70710678118654752f));
    }
  }

  // ---- Direct GEMM: each wave owns 96 columns x 32 rows = 12 C tiles, K = 768.
  //      Each B fragment (from L2-resident W) feeds two WMMAs (both M-subtiles). ----
  const int nbase = wid * 96;
  v8f acc[2][6];
  #pragma unroll
  for (int s = 0; s < 2; ++s)
    #pragma unroll
    for (int t = 0; t < 6; ++t) acc[s][t] = (v8f){};
  const float* wrow[6];
  #pragma unroll
  for (int t = 0; t < 6; ++t) wrow[t] = W + (size_t)(nbase + t * 16 + lm) * DIN;
  {
    const float* arow0 = &xs[lm * XS_LD];          // rows 0..15
    const float* arow1 = &xs[(16 + lm) * XS_LD];   // rows 16..31
    for (int k = 0; k < DIN; k += 4) {
      v2f a0 = *(const v2f*)&arow0[k + 2 * half];
      v2f a1 = *(const v2f*)&arow1[k + 2 * half];
      #pragma unroll
      for (int t = 0; t < 6; ++t) {
        v2f bb = *(const v2f*)&wrow[t][k + 2 * half];
        acc[0][t] = wmma4(a0, bb, acc[0][t]);
        acc[1][t] = wmma4(a1, bb, acc[1][t]);
      }
    }
  }
  __syncthreads();   // wavs published

  // ---- wpath GEMM: same accumulators, A = wavs, B = Wc, K = 48 ----
  {
    const float* arow0 = &wavs[lm * WV_LD];
    const float* arow1 = &wavs[(16 + lm) * WV_LD];
    const float* wcrow[6];
    #pragma unroll
    for (int t = 0; t < 6; ++t) wcrow[t] = Wc + (size_t)(nbase + t * 16 + lm) * NW;
    for (int k = 0; k < NW; k += 4) {
      v2f a0 = *(const v2f*)&arow0[k + 2 * half];
      v2f a1 = *(const v2f*)&arow1[k + 2 * half];
      #pragma unroll
      for (int t = 0; t < 6; ++t) {
        v2f bb = *(const v2f*)&wcrow[t][k + 2 * half];
        acc[0][t] = wmma4(a0, bb, acc[0][t]);
        acc[1][t] = wmma4(a1, bb, acc[1][t]);
      }
    }
  }

  // ---- bias + per-row LayerNorm stats (deterministic tree reduction) ----
  float sum16[2][8], sq16[2][8];
  #pragma unroll
  for (int s = 0; s < 2; ++s)
    #pragma unroll
    for (int r = 0; r < 8; ++r) { sum16[s][r] = 0.f; sq16[s][r] = 0.f; }
  #pragma unroll
  for (int t = 0; t < 6; ++t) {
    int   n    = nbase + t * 16 + lm;
    float bias = b[n] + bc[n];
    #pragma unroll
    for (int s = 0; s < 2; ++s) {
      #pragma unroll
      for (int r = 0; r < 8; ++r) {
        float v = acc[s][t][r] + bias;
        acc[s][t][r] = v;
        sum16[s][r] += v;
        sq16[s][r]  += v * v;
      }
    }
  }
  #pragma unroll
  for (int off = 1; off < 16; off <<= 1) {  // xor-butterfly within each 16-lane half
    #pragma unroll
    for (int s = 0; s < 2; ++s) {
      #pragma unroll
      for (int r = 0; r < 8; ++r) {
        sum16[s][r] += __shfl_xor(sum16[s][r], off, 32);
        sq16[s][r]  += __shfl_xor(sq16[s][r],  off, 32);
      }
    }
  }
  if (lm == 0) {
    #pragma unroll
    for (int s = 0; s < 2; ++s) {
      #pragma unroll
      for (int r = 0; r < 8; ++r) {
        int m = r + 8 * half + 16 * s;
        psum[wid][m] = sum16[s][r];
        psq[wid][m]  = sq16[s][r];
      }
    }
  }
  __syncthreads();
  if (tid < M_TILE) {
    float s = 0.f, q = 0.f;
    #pragma unroll
    for (int w = 0; w < 8; ++w) { s += psum[w][tid]; q += psq[w][tid]; }
    float mu  = s * (1.0f / DOUT);
    float var = q * (1.0f / DOUT) - mu * mu;
    s_mu[tid]  = mu;
    s_inv[tid] = rsqrtf(var + LN_EPS);
  }
  __syncthreads();

  // ---- normalize + store (NT stores: out is written once, keep L2 for W) ----
  float mu16[2][8], inv16[2][8];
  #pragma unroll
  for (int s = 0; s < 2; ++s)
    #pragma unroll
    for (int r = 0; r < 8; ++r) {
      int m = r + 8 * half + 16 * s;
      mu16[s][r]  = s_mu[m];
      inv16[s][r] = s_inv[m];
    }
  #pragma unroll
  for (int t = 0; t < 6; ++t) {
    int   n  = nbase + t * 16 + lm;
    float g  = gamma[n];
    float bt = beta[n];
    #pragma unroll
    for (int s = 0; s < 2; ++s) {
      #pragma unroll
      for (int r = 0; r < 8; ++r) {
        int m = r + 8 * half + 16 * s;
        float v = (acc[s][t][r] - mu16[s][r]) * inv16[s][r] * g + bt;
        __builtin_nontemporal_store(v, out + (size_t)(m0 + m) * DOUT + n);
      }
    }
  }
}

extern "C" void kernel_launch(void* const* d_in, const int* in_sizes, int n_in,
                              void* d_out, int out_size, void* d_ws, size_t ws_size,
                              hipStream_t stream) {
  (void)in_sizes; (void)n_in; (void)d_ws; (void)ws_size; (void)out_size;
  const float* x      = (const float*)d_in[0];
  const float* W      = (const float*)d_in[1];
  const float* b      = (const float*)d_in[2];
  const float* Wp     = (const float*)d_in[3];
  const float* bp     = (const float*)d_in[4];
  const float* Wc     = (const float*)d_in[5];
  const float* bc     = (const float*)d_in[6];
  const float* scales = (const float*)d_in[7];
  const float* trans  = (const float*)d_in[8];
  const float* gamma  = (const float*)d_in[9];
  const float* beta   = (const float*)d_in[10];
  float* out = (float*)d_out;

  dim3 grid(NROWS / M_TILE);   // 1024 blocks
  dim3 block(256);             // 8 waves (wave32)
  wkan_fused_kernel<<<grid, block, 0, stream>>>(x, W, b, Wp, bp, Wc, bc,
                                                scales, trans, gamma, beta, out);
}